// SimpleCPNN_13529146982626
// MI455X (gfx1250) — compile-verified
//
#include <hip/hip_runtime.h>

// ---------------------------------------------------------------------------
// CPNN forward for MI455X (gfx1250, wave32):
//   winners[b] = argmin_h ( ||w_h||^2 - 2 * x_b . w_h )   [ ||x_b||^2 dropped ]
//   out[b,:]   = G[:, winners[b]]
// Core GEMM: V_WMMA_F32_16X16X4_F32 (exact f32 math -> exact winners).
// x block staged into LDS with GLOBAL_LOAD_ASYNC_TO_LDS_B128 (ASYNCcnt).
// ---------------------------------------------------------------------------

typedef __attribute__((ext_vector_type(2))) float v2f;
typedef __attribute__((ext_vector_type(8))) float v8f;

#define B_   4096
#define D_   1024
#define H_   8192
#define O_   1000

#define BM   32            // x rows per workgroup (2 WMMA M sub-tiles)
#define HT   1024          // H columns per workgroup (grid.y slice)
#define HB   (H_ / HT)     // 8 partials per row
#define XSTR 1028          // LDS row stride (floats): 1024+4 pad -> conflict-free A-frag ds_load_b64
                           // row stride bytes = 4112 (16B aligned for b128 async copies)

// ---------------- kernel 1: w_sq[h] = sum_d W[h][d]^2 (one wave per row) ----
__global__ __launch_bounds__(256) void wsq_kernel(const float* __restrict__ W,
                                                  float* __restrict__ wsq) {
    int h    = blockIdx.x * 8 + (threadIdx.x >> 5);
    int lane = threadIdx.x & 31;
    const float4* row = reinterpret_cast<const float4*>(W + (size_t)h * D_);
    float s = 0.f;
#pragma unroll
    for (int i = 0; i < (D_ / 4) / 32; ++i) {
        float4 v = row[lane + i * 32];
        s = fmaf(v.x, v.x, fmaf(v.y, v.y, fmaf(v.z, v.z, fmaf(v.w, v.w, s))));
    }
#pragma unroll
    for (int off = 16; off > 0; off >>= 1) s += __shfl_xor(s, off, 32);
    if (lane == 0) wsq[h] = s;
}

// ---------------- kernel 2: fused WMMA distance GEMM + per-slice argmin -----
__global__ __launch_bounds__(256) void dist_argmin_kernel(
        const float* __restrict__ x, const float* __restrict__ W,
        const float* __restrict__ wsq,
        float* __restrict__ pval, int* __restrict__ pidx) {
    __shared__ float xs[BM][XSTR];        // ~131.6 KB of 320 KB WGP LDS
    __shared__ float redv[BM][8];
    __shared__ int   redi[BM][8];

    const int bx = blockIdx.x, by = blockIdx.y;
    const int tid = threadIdx.x;

    // ---- Stage x block [BM][D_] into LDS with async copies (ASYNCcnt). ----
    {
        const char* gbase = (const char*)(x + (size_t)bx * BM * D_);
        unsigned ldsbase = (unsigned)(uintptr_t)&xs[0][0];   // low 32 bits = LDS offset
#pragma unroll
        for (int j = 0; j < (BM * D_) / (4 * 256); ++j) {    // 32 x b128 per thread
            int i = tid + j * 256;
            int r = i >> 8;                                  // 256 x 16B per row
            int c = i & 255;
            unsigned lds_addr = ldsbase + (unsigned)(r * (XSTR * 4) + c * 16);
            const char* gaddr = gbase + (size_t)r * (D_ * 4) + (size_t)c * 16;
            asm volatile("global_load_async_to_lds_b128 %0, %1, off"
                         :: "v"(lds_addr), "v"(gaddr) : "memory");
        }
#if __has_builtin(__builtin_amdgcn_s_wait_asynccnt)
        __builtin_amdgcn_s_wait_asynccnt(0);
#else
        asm volatile("s_wait_asynccnt 0x0" ::: "memory");
#endif
    }
    __syncthreads();

    const int wave = tid >> 5, lane = tid & 31;   // wave = N-wave index (0..7)
    const int li = lane & 15, lh = lane >> 4;

    float mval[16];
    int   midx[16];
#pragma unroll
    for (int r = 0; r < 16; ++r) { mval[r] = __builtin_inff(); midx[r] = 0x7fffffff; }

    // Each wave: 8 column-tiles of 16 cols; both M sub-tiles; full K=1024 per tile.
    for (int t = 0; t < HT / (16 * 8); ++t) {
        const int col = by * HT + wave * 16 + t * 128 + li;     // this lane's column (N = li)
        const float* bptr = W + (size_t)col * D_ + 2 * lh;      // B-frag: W row 'col', K pair
        v8f acc0 = {};                                          // rows 0..15
        v8f acc1 = {};                                          // rows 16..31
#pragma unroll 8
        for (int k = 0; k < D_; k += 4) {
            // A-frags (ISA 16x4 layout): lane L -> M = L%16, K = k + 2*(L>=16) .. +1
            v2f a0 = *reinterpret_cast<const v2f*>(&xs[li][2 * lh + k]);
            v2f a1 = *reinterpret_cast<const v2f*>(&xs[16 + li][2 * lh + k]);
            // B-frag (4x16 layout): lane L -> N = L%16, K = k + 2*(L>=16) .. +1
            v2f b = *reinterpret_cast<const v2f*>(bptr + k);
            acc0 = __builtin_amdgcn_wmma_f32_16x16x4_f32(false, a0, false, b,
                                                         (short)0, acc0, false, false);
            acc1 = __builtin_amdgcn_wmma_f32_16x16x4_f32(false, a1, false, b,
                                                         (short)0, acc1, false, false);
        }
        const float wq = wsq[col];
#pragma unroll
        for (int r = 0; r < 8; ++r) {
            // C layout: VGPR r, lane L -> row (sub-tile base) + r + 8*lh, col 'col'
            float s0 = fmaf(-2.f, acc0[r], wq);
            float s1 = fmaf(-2.f, acc1[r], wq);
            if (s0 < mval[r])     { mval[r]     = s0; midx[r]     = col; }
            if (s1 < mval[8 + r]) { mval[8 + r] = s1; midx[8 + r] = col; }
        }
    }

    // Cross-lane argmin within each 16-lane half (same rows, different columns).
#pragma unroll
    for (int off = 1; off < 16; off <<= 1) {
#pragma unroll
        for (int r = 0; r < 16; ++r) {
            float ov = __shfl_xor(mval[r], off, 32);
            int   oi = __shfl_xor(midx[r], off, 32);
            if (ov < mval[r] || (ov == mval[r] && oi < midx[r])) { mval[r] = ov; midx[r] = oi; }
        }
    }
    if (li == 0) {
#pragma unroll
        for (int r = 0; r < 8; ++r) {
            int row0 = 8 * lh + r;            // rows 0..15
            int row1 = 16 + 8 * lh + r;       // rows 16..31
            redv[row0][wave] = mval[r];      redi[row0][wave] = midx[r];
            redv[row1][wave] = mval[8 + r];  redi[row1][wave] = midx[8 + r];
        }
    }
    __syncthreads();

    // Combine the 8 N-waves; emit one partial (min, argmin) per row per H-slice.
    if (tid < BM) {
        float bv = redv[tid][0]; int bi = redi[tid][0];
#pragma unroll
        for (int j = 1; j < 8; ++j) {
            float v = redv[tid][j]; int i2 = redi[tid][j];
            if (v < bv || (v == bv && i2 < bi)) { bv = v; bi = i2; }
        }
        size_t rowg = (size_t)bx * BM + tid;
        pval[rowg * HB + by] = bv;
        pidx[rowg * HB + by] = bi;
    }
}

// ---------------- kernel 3: reduce the HB partials per row ------------------
__global__ __launch_bounds__(256) void reduce_winners_kernel(
        const float* __restrict__ pval, const int* __restrict__ pidx,
        int* __restrict__ winners) {
    int b = blockIdx.x * 256 + threadIdx.x;
    if (b >= B_) return;
    float bv = pval[(size_t)b * HB]; int bi = pidx[(size_t)b * HB];
#pragma unroll
    for (int j = 1; j < HB; ++j) {
        float v = pval[(size_t)b * HB + j]; int i2 = pidx[(size_t)b * HB + j];
        if (v < bv || (v == bv && i2 < bi)) { bv = v; bi = i2; }
    }
    winners[b] = bi;
}

// ---------------- kernel 4: out[b,:] = G[:, winners[b]]; emit winners -------
__global__ __launch_bounds__(256) void gather_kernel(
        const float* __restrict__ G, const int* __restrict__ winners,
        float* __restrict__ out, float* __restrict__ wout) {
    int b = blockIdx.x;
    int w = winners[b];
    for (int o = threadIdx.x; o < O_; o += 256)
        out[(size_t)b * O_ + o] = G[(size_t)o * H_ + w];
    if (threadIdx.x == 0) wout[b] = (float)w;
}

// ---------------------------------------------------------------------------
extern "C" void kernel_launch(void* const* d_in, const int* in_sizes, int n_in,
                              void* d_out, int out_size, void* d_ws, size_t ws_size,
                              hipStream_t stream) {
    (void)in_sizes; (void)n_in; (void)out_size; (void)ws_size;
    const float* x  = (const float*)d_in[0];   // [B, D]
    const float* Wk = (const float*)d_in[1];   // [H, D]
    const float* G  = (const float*)d_in[2];   // [O, H]
    float* out = (float*)d_out;                // [B*O] floats, then [B] winners-as-float

    char* ws = (char*)d_ws;
    float* wsq  = (float*)(ws);                               // 32 KB
    float* pval = (float*)(ws + 32 * 1024);                   // 128 KB
    int*   pidx = (int*)  (ws + 32 * 1024 + 128 * 1024);      // 128 KB
    int*   win  = (int*)  (ws + 32 * 1024 + 256 * 1024);      // 16 KB

    wsq_kernel<<<H_ / 8, 256, 0, stream>>>(Wk, wsq);

    dim3 g2(B_ / BM, HB);
    dist_argmin_kernel<<<g2, 256, 0, stream>>>(x, Wk, wsq, pval, pidx);

    reduce_winners_kernel<<<B_ / 256, 256, 0, stream>>>(pval, pidx, win);

    gather_kernel<<<B_, 256, 0, stream>>>(G, win, out, out + (size_t)B_ * O_);
}